// FanInMP_2972117369426
// MI455X (gfx1250) — compile-verified
//
#include <hip/hip_runtime.h>
#include <hip/hip_bf16.h>
#include <math.h>
#include <stdint.h>

// ---------------------------------------------------------------------------
// FanInMP segment-logsumexp for MI455X (gfx1250, wave32).
// CSR-invert the edge list (count -> atomic reserve -> fill), then one wave
// per destination row streams its gathered 512B tiles through LDS with a
// depth-8 async pipeline (global_load_async_to_lds_b128 / s_wait_asynccnt),
// doing an online logsumexp entirely in registers. Single pass over x_src,
// zero float atomics.
// ---------------------------------------------------------------------------

#define WPB  8   // waves (dst rows) per block in the main kernel
#define PIPE 8   // async tiles in flight per wave (ring of 8 x 512B in LDS)

typedef float nfloat4 __attribute__((ext_vector_type(4)));  // native vec4

__device__ __forceinline__ void async_load_b128(const float* g, float* l) {
  // `l` points into __shared__. Per CDNA5 ISA §10.2 (aperture mapping), the
  // low 32 bits of a flat LDS-aperture address are the LDS byte offset, which
  // is exactly what the async instruction's VDST operand wants.
  unsigned lofs = (unsigned)(uintptr_t)l;
  asm volatile("global_load_async_to_lds_b128 %0, %1, off"
               :: "v"(lofs), "v"(g) : "memory");
}

template <int N>
__device__ __forceinline__ void wait_asynccnt() {
  // asm (not builtin) so the "memory" clobber is a hard compiler barrier:
  // the LDS tile read cannot be hoisted above the wait, and async issues
  // cannot sink below it.
  asm volatile("s_wait_asynccnt %0" :: "i"(N) : "memory");
}

__device__ __forceinline__ void wait_for_rem(int rem) {
  if (rem >= 7)      wait_asynccnt<7>();
  else if (rem == 6) wait_asynccnt<6>();
  else if (rem == 5) wait_asynccnt<5>();
  else if (rem == 4) wait_asynccnt<4>();
  else if (rem == 3) wait_asynccnt<3>();
  else if (rem == 2) wait_asynccnt<2>();
  else if (rem == 1) wait_asynccnt<1>();
  else               wait_asynccnt<0>();
}

__device__ __forceinline__ void lse_update(nfloat4& m, nfloat4& s, const nfloat4 v) {
  float t;
  t = fmaxf(m.x, v.x); s.x = s.x * __expf(m.x - t) + __expf(v.x - t); m.x = t;
  t = fmaxf(m.y, v.y); s.y = s.y * __expf(m.y - t) + __expf(v.y - t); m.y = t;
  t = fmaxf(m.z, v.z); s.z = s.z * __expf(m.z - t) + __expf(v.z - t); m.z = t;
  t = fmaxf(m.w, v.w); s.w = s.w * __expf(m.w - t) + __expf(v.w - t); m.w = t;
}

// One wave per destination row. Lane l owns channels [4l, 4l+3].
__global__ void __launch_bounds__(WPB * 32)
k_lse(const float* __restrict__ x, const int* __restrict__ lst,
      const int* __restrict__ base, const int* __restrict__ cnt,
      float* __restrict__ out, int ndst) {
  __shared__ float buf[WPB][PIPE][128];  // 32KB: depth-8 512B tile ring
  const int lane  = threadIdx.x & 31;
  const int wslot = threadIdx.x >> 5;
  const int dst   = blockIdx.x * WPB + wslot;
  if (dst >= ndst) return;

  // dst (hence beg/n) is wave-uniform; tell the compiler so the path split
  // and loop control go through SALU (s_cmp/s_cbranch) instead of EXEC masks.
  const int beg = __builtin_amdgcn_readfirstlane(base[dst]);
  const int n   = __builtin_amdgcn_readfirstlane(cnt[dst]);

  nfloat4 m = {-INFINITY, -INFINITY, -INFINITY, -INFINITY};
  nfloat4 s = {0.f, 0.f, 0.f, 0.f};
  float* const lb = &buf[wslot][0][0];  // wave's ring base; slot k at lb + k*128

  if (n > 0 && n <= 32) {
    // Fast path (covers ~all rows, avg segment = 16): fetch every list offset
    // in ONE coalesced lane-parallel load, broadcast via shuffle, keep up to
    // PIPE gathers in flight so issue rate is never dependent-load bound.
    const int myoff = (lane < n) ? lst[beg + lane] : 0;
#pragma unroll
    for (int k = 0; k < PIPE - 1; ++k)
      if (k < n)
        async_load_b128(x + __shfl(myoff, k) + lane * 4, lb + k * 128 + lane * 4);
    for (int i = 0; i < n; ++i) {
      if (i + PIPE - 1 < n)
        async_load_b128(x + __shfl(myoff, i + PIPE - 1) + lane * 4,
                        lb + ((i + PIPE - 1) & (PIPE - 1)) * 128 + lane * 4);
      wait_for_rem(n - 1 - i);  // tiles newer than the one we consume
      const nfloat4 v = *reinterpret_cast<const nfloat4*>(
          lb + (i & (PIPE - 1)) * 128 + lane * 4);
      lse_update(m, s, v);
    }
  } else if (n > 32) {
    // Rare long segments (P[n>32] ~ 2e-4): simple double-buffered pipeline.
    async_load_b128(x + lst[beg] + lane * 4, lb + lane * 4);
    int par = 0;
    for (int i = 0; i < n; ++i) {
      const bool more = (i + 1 < n);
      if (more)
        async_load_b128(x + lst[beg + i + 1] + lane * 4,
                        lb + (par ^ 1) * 128 + lane * 4);
      if (more) wait_asynccnt<1>(); else wait_asynccnt<0>();
      const nfloat4 v =
          *reinterpret_cast<const nfloat4*>(lb + par * 128 + lane * 4);
      lse_update(m, s, v);
      par ^= 1;
    }
  }

  nfloat4 o;
  o.x = (n > 0) ? m.x + __logf(s.x) : 0.f;  // s >= 1 when n > 0 (max term = e^0)
  o.y = (n > 0) ? m.y + __logf(s.y) : 0.f;
  o.z = (n > 0) ? m.z + __logf(s.z) : 0.f;
  o.w = (n > 0) ? m.w + __logf(s.w) : 0.f;
  // Write-once output: non-temporal so it doesn't evict x_src / CSR from L2.
  __builtin_nontemporal_store(o, reinterpret_cast<nfloat4*>(
      out + (size_t)dst * 128 + lane * 4));
}

// ---------------------------------------------------------------------------

__global__ void k_zero_i32(int* __restrict__ p, int n) {
  int i = blockIdx.x * blockDim.x + threadIdx.x;
  if (i < n) p[i] = 0;
}

__global__ void k_count(const long long* __restrict__ edge_dst,
                        int* __restrict__ cnt, int ne) {
  int i = blockIdx.x * blockDim.x + threadIdx.x;
  if (i < ne) atomicAdd(&cnt[(int)__builtin_nontemporal_load(&edge_dst[i])], 1);
}

// Segment placement via atomic reserve: base[dst] = fetch_add(total, cnt[dst]).
// Uniform address -> AMDGPU atomic optimizer turns this into a wave-local scan
// plus ONE atomic per wave (~3K serialized L2 atomics total, microseconds).
// Placement order is irrelevant: each dst still gets a private contiguous range.
__global__ void k_reserve(const int* __restrict__ cnt, int* __restrict__ base,
                          unsigned* __restrict__ total, int n) {
  int i = blockIdx.x * blockDim.x + threadIdx.x;
  if (i < n) base[i] = (int)atomicAdd(total, (unsigned)cnt[i]);
}

__global__ void k_fill(const long long* __restrict__ edge_src,
                       const long long* __restrict__ edge_dst,
                       const int* __restrict__ base, int* __restrict__ cur,
                       int* __restrict__ lst, int ne, int ept) {
  int i = blockIdx.x * blockDim.x + threadIdx.x;
  if (i >= ne) return;
  const int dst = (int)__builtin_nontemporal_load(&edge_dst[i]);
  const int src = (int)__builtin_nontemporal_load(&edge_src[i]);
  const int p   = i / ept;  // port index (row-major (P,EPT) flattening)
  const int pos = base[dst] + atomicAdd(&cur[dst], 1);
  lst[pos] = src * 512 + p * 128;  // float offset of the 128-wide message row
}

// ---------------------------------------------------------------------------

extern "C" void kernel_launch(void* const* d_in, const int* in_sizes, int n_in,
                              void* d_out, int out_size, void* d_ws, size_t ws_size,
                              hipStream_t stream) {
  const float*     x    = (const float*)d_in[0];
  const long long* esrc = (const long long*)d_in[1];  // int64 (P, EPT)
  const long long* edst = (const long long*)d_in[2];  // int64 (P, EPT)
  float*           out  = (float*)d_out;

  const int NE   = in_sizes[1];     // 1,600,000 edges total
  const int NDST = out_size / 128;  // 100,000
  const int EPT  = NE / 4;          // 400,000 edges per port

  // Workspace: cnt | cur | total | base | list  ~= 7.6 MB
  int*      cnt   = (int*)d_ws;
  int*      cur   = cnt + NDST;
  unsigned* total = (unsigned*)(cur + NDST);
  int*      base  = (int*)(total + 1);
  int*      lst   = base + NDST;

  k_zero_i32<<<(2 * NDST + 1 + 255) / 256, 256, 0, stream>>>(cnt, 2 * NDST + 1);
  k_count   <<<(NE + 255) / 256, 256, 0, stream>>>(edst, cnt, NE);
  k_reserve <<<(NDST + 255) / 256, 256, 0, stream>>>(cnt, base, total, NDST);
  k_fill    <<<(NE + 255) / 256, 256, 0, stream>>>(esrc, edst, base, cur, lst, NE, EPT);
  k_lse     <<<(NDST + WPB - 1) / WPB, WPB * 32, 0, stream>>>(x, lst, base, cnt, out, NDST);
}